// AttentionModel_43963285242805
// MI455X (gfx1250) — compile-verified
//
#include <hip/hip_runtime.h>
#include <math.h>

// Problem constants (from reference): B=512, N=128, D=128, H=8, DK=16
constexpr int kB  = 512;
constexpr int kN  = 128;
constexpr int kD  = 128;
constexpr int kH  = 8;
constexpr int kDK = 16;
constexpr int kTD = 384;          // 3*D  (gK | gV | lK packed per node row)
constexpr int kBPG = 16;          // batch elements per workgroup in decode kernel

typedef float v2f __attribute__((ext_vector_type(2)));
typedef float v8f __attribute__((ext_vector_type(8)));

// fp32 WMMA, D = A(16x4) * B(4x16) + C(16x16)   (ISA V_WMMA_F32_16X16X4_F32)
__device__ __forceinline__ v8f wmma4(v2f a, v2f b, v8f c) {
  return __builtin_amdgcn_wmma_f32_16x16x4_f32(false, a, false, b, (short)0, c,
                                               false, false);
}

__device__ __forceinline__ float dot4(float4 a, float4 b) {
  return a.x * b.x + a.y * b.y + a.z * b.z + a.w * b.w;
}

// ---------------------------------------------------------------------------
// Kernel 1: QKV = h @ W_node   (M = B*N = 65536, K = 128, N = 384)
// One WG = 128 threads = 4 waves; WG owns a 16-row M tile; each wave does 6
// 16-wide N tiles; K-loop of 32 fp32 WMMAs per tile.
// ---------------------------------------------------------------------------
__global__ void qkv_kernel(const float* __restrict__ h,
                           const float* __restrict__ Wn,
                           float* __restrict__ qkv) {
  __shared__ float As[16][132];                 // 16x128 A tile (+pad)
  const int m0  = blockIdx.x * 16;
  const int tid = threadIdx.x;
  for (int i = tid; i < 512; i += 128) {        // 512 float4 = 16x128 floats
    const int r = i >> 5, c4 = i & 31;
    *(float4*)&As[r][c4 * 4] = ((const float4*)&h[(size_t)(m0 + r) * kD])[c4];
  }
  __syncthreads();

  const int wave  = tid >> 5;
  const int lane  = tid & 31;
  const int lrow  = lane & 15;
  const int khalf = (lane >> 4) * 2;            // A/B: upper half-wave holds K+2,K+3
  const int rbase = (lane >> 4) * 8;            // C/D: upper half-wave holds M+8

  for (int jt = wave * 6; jt < wave * 6 + 6; ++jt) {
    const int col = jt * 16 + lrow;
    v8f c = {0.f, 0.f, 0.f, 0.f, 0.f, 0.f, 0.f, 0.f};
    for (int ks = 0; ks < 32; ++ks) {
      const int k = ks * 4 + khalf;
      v2f a; a.x = As[lrow][k];               a.y = As[lrow][k + 1];
      v2f b; b.x = Wn[(size_t)k * kTD + col]; b.y = Wn[(size_t)(k + 1) * kTD + col];
      c = wmma4(a, b, c);
    }
#pragma unroll
    for (int j = 0; j < 8; ++j)
      qkv[(size_t)(m0 + rbase + j) * kTD + col] = c[j];
  }
}

// ---------------------------------------------------------------------------
// Kernel 2: fixed_ctx = mean_n(h) @ W_fixed     (tiny: VALU, 1 WG per b)
// ---------------------------------------------------------------------------
__global__ void fixed_kernel(const float* __restrict__ h,
                             const float* __restrict__ Wf,
                             float* __restrict__ fixedc) {
  __shared__ float ge[kD];
  const int b = blockIdx.x, t = threadIdx.x;
  float s = 0.f;
  for (int n = 0; n < kN; ++n) s += h[((size_t)b * kN + n) * kD + t];
  ge[t] = s * (1.0f / kN);
  __syncthreads();
  float f = 0.f;
  for (int d = 0; d < kD; ++d) f += ge[d] * Wf[d * kD + t];
  fixedc[b * kD + t] = f;
}

// ---------------------------------------------------------------------------
// Kernel 3: FUSED 128-step greedy decode. One launch; 32 WGs x 512 threads
// (16 waves). Each WG exclusively owns 16 batch elements -> the whole
// autoregressive loop needs only workgroup barriers (no global sync, no
// relaunch); decode state lives in LDS for all 128 steps. WMMA (fp32
// 16x16x4, waves 0-7) for q = fixed + ctx@W_step and glimpse = heads@W_out;
// one wave per batch element for the masked softmax / log-softmax / argmax
// stages, with b128 (float4) global+LDS accesses on every hot dot product.
// ---------------------------------------------------------------------------
__global__ void decode_kernel(const float* __restrict__ h,
                              const float* __restrict__ Wp,
                              const float* __restrict__ Ws,
                              const float* __restrict__ Wo,
                              const float* __restrict__ qkv,
                              const float* __restrict__ fixedc,
                              float* __restrict__ out_logp,
                              float* __restrict__ out_seq) {
  __shared__ float ctx[kBPG][260];       // step context  [16][256]+pad
  __shared__ float qbuf[kBPG][132];      // q per batch row
  __shared__ float hb[kBPG][132];        // concatenated attention heads
  __shared__ float gl[kBPG][132];        // glimpse
  __shared__ float fxl[kBPG][132];       // cached fixed_ctx rows
  __shared__ float attnw[16][kN];        // per-wave softmax scratch
  __shared__ unsigned vis[kBPG][4];      // visited bitmask per b
  __shared__ int firstL[kBPG], prevL[kBPG];

  const int b0   = blockIdx.x * kBPG;
  const int tid  = threadIdx.x;          // 0..511
  const int wave = tid >> 5, lane = tid & 31;
  const int lrow  = lane & 15;
  const int khalf = (lane >> 4) * 2;     // A/B frag: upper half-wave holds K+2,K+3
  const int rbase = (lane >> 4) * 8;     // C/D frag: upper half-wave holds M+8

  // ---- one-time init: cache fixed_ctx in LDS, zero decode state
  for (int i = tid; i < kBPG * kD; i += 512)
    fxl[i >> 7][i & 127] = fixedc[(b0 + (i >> 7)) * kD + (i & 127)];
  if (tid < kBPG) {
    vis[tid][0] = vis[tid][1] = vis[tid][2] = vis[tid][3] = 0u;
    firstL[tid] = 0; prevL[tid] = 0;
  }
  __syncthreads();

  for (int step = 0; step < kN; ++step) {
    // ---- stage 1: ctx rows (placeholder on step 0, else h[first] || h[prev])
    for (int i = tid; i < kBPG * 64; i += 512) {   // 64 float4 per 256-f row
      const int r = i >> 6, c4 = i & 63;
      float4 v;
      if (step == 0) v = ((const float4*)Wp)[c4];
      else {
        const int src = (c4 < 32) ? firstL[r] : prevL[r];
        v = ((const float4*)&h[((size_t)(b0 + r) * kN + src) * kD])[c4 & 31];
      }
      *(float4*)&ctx[r][c4 * 4] = v;
    }
    __syncthreads();

    // ---- stage 2: q = fixed_ctx + ctx @ W_step  (M=16,K=256,N=128) — WMMA
    if (wave < 8) {
      const int col = wave * 16 + lrow;            // each wave owns one N tile
      v8f c;
#pragma unroll
      for (int j = 0; j < 8; ++j) c[j] = fxl[rbase + j][col];
      for (int ks = 0; ks < 64; ++ks) {
        const int k = ks * 4 + khalf;
        v2f a;  a.x = ctx[lrow][k];        a.y = ctx[lrow][k + 1];
        v2f bb; bb.x = Ws[k * kD + col];   bb.y = Ws[(k + 1) * kD + col];
        c = wmma4(a, bb, c);
      }
#pragma unroll
      for (int j = 0; j < 8; ++j) qbuf[rbase + j][col] = c[j];
    }
    __syncthreads();

    // ---- stage 3: masked multi-head attention (one wave per batch element)
    {
      const int r = wave;
      const int b = b0 + r;
      unsigned vw[4];
#pragma unroll
      for (int t = 0; t < 4; ++t) vw[t] = vis[r][t];   // n=32t+lane: word t, bit lane

      for (int hh = 0; hh < kH; ++hh) {
        const float4 q0 = *(const float4*)&qbuf[r][hh * kDK + 0];
        const float4 q1 = *(const float4*)&qbuf[r][hh * kDK + 4];
        const float4 q2 = *(const float4*)&qbuf[r][hh * kDK + 8];
        const float4 q3 = *(const float4*)&qbuf[r][hh * kDK + 12];

        float cv[4], m = -INFINITY;
#pragma unroll
        for (int t = 0; t < 4; ++t) {
          const int n = t * 32 + lane;
          const float4* kr4 =
              (const float4*)&qkv[((size_t)b * kN + n) * kTD + hh * kDK];
          float s = dot4(q0, kr4[0]) + dot4(q1, kr4[1]) +
                    dot4(q2, kr4[2]) + dot4(q3, kr4[3]);
          s *= 0.25f;                                  // 1/sqrt(DK)
          if ((vw[t] >> lane) & 1u) s = -INFINITY;
          cv[t] = s; m = fmaxf(m, s);
        }
        for (int off = 16; off > 0; off >>= 1) m = fmaxf(m, __shfl_xor(m, off, 32));
        float ssum = 0.f, ev[4];
#pragma unroll
        for (int t = 0; t < 4; ++t) { ev[t] = __expf(cv[t] - m); ssum += ev[t]; }
        for (int off = 16; off > 0; off >>= 1) ssum += __shfl_xor(ssum, off, 32);
        const float inv = 1.0f / ssum;
#pragma unroll
        for (int t = 0; t < 4; ++t) attnw[wave][t * 32 + lane] = ev[t] * inv;

        // heads[hh][k] = sum_n attn[n]*gV[b][n][hh*16+k].
        // Lane owns 4 consecutive k (one float4/row); 8 lane-groups split n.
        const int k4 = (lane & 3) * 4;
        const int n0 = (lane >> 2) * 16;
        float4 acc = {0.f, 0.f, 0.f, 0.f};
        for (int n = n0; n < n0 + 16; ++n) {
          const float a = attnw[wave][n];
          const float4 v =
              *(const float4*)&qkv[((size_t)b * kN + n) * kTD + kD + hh * kDK + k4];
          acc.x += a * v.x; acc.y += a * v.y; acc.z += a * v.z; acc.w += a * v.w;
        }
#pragma unroll
        for (int off = 4; off <= 16; off <<= 1) {
          acc.x += __shfl_xor(acc.x, off, 32);
          acc.y += __shfl_xor(acc.y, off, 32);
          acc.z += __shfl_xor(acc.z, off, 32);
          acc.w += __shfl_xor(acc.w, off, 32);
        }
        if ((lane >> 2) == 0)
          *(float4*)&hb[r][hh * kDK + k4] = acc;
      }
    }
    __syncthreads();

    // ---- stage 4: glimpse = heads @ W_out  (M=16,K=128,N=128) — WMMA
    if (wave < 8) {
      const int col = wave * 16 + lrow;
      v8f c = {0.f, 0.f, 0.f, 0.f, 0.f, 0.f, 0.f, 0.f};
      for (int ks = 0; ks < 32; ++ks) {
        const int k = ks * 4 + khalf;
        v2f a;  a.x = hb[lrow][k];        a.y = hb[lrow][k + 1];
        v2f bb; bb.x = Wo[k * kD + col];  bb.y = Wo[(k + 1) * kD + col];
        c = wmma4(a, bb, c);
      }
#pragma unroll
      for (int j = 0; j < 8; ++j) gl[rbase + j][col] = c[j];
    }
    __syncthreads();

    // ---- stage 5: tanh-clipped logits, log-softmax, greedy argmax, update
    {
      const int r = wave;
      const int b = b0 + r;
      unsigned vw[4];
#pragma unroll
      for (int t = 0; t < 4; ++t) vw[t] = vis[r][t];
      const float4* g4 = (const float4*)&gl[r][0];

      float lv[4], m = -INFINITY;
#pragma unroll
      for (int t = 0; t < 4; ++t) {
        const int n = t * 32 + lane;
        const float4* lk4 =
            (const float4*)&qkv[((size_t)b * kN + n) * kTD + 2 * kD];
        float s = 0.f;
#pragma unroll 8
        for (int d4 = 0; d4 < 32; ++d4) s += dot4(g4[d4], lk4[d4]);
        s = 10.0f * tanhf(s * 0.08838834764831845f);   // tanh clip, 1/sqrt(D)
        if ((vw[t] >> lane) & 1u) s = -INFINITY;
        lv[t] = s; m = fmaxf(m, s);
      }
      for (int off = 16; off > 0; off >>= 1) m = fmaxf(m, __shfl_xor(m, off, 32));
      float ssum = 0.f;
#pragma unroll
      for (int t = 0; t < 4; ++t) ssum += __expf(lv[t] - m);
      for (int off = 16; off > 0; off >>= 1) ssum += __shfl_xor(ssum, off, 32);
      const float lse = m + __logf(ssum);

      float bestv = -INFINITY; int besti = kN;         // strict > keeps first max
#pragma unroll
      for (int t = 0; t < 4; ++t) {
        const int n = t * 32 + lane;
        out_logp[((size_t)b * kN + step) * kN + n] = lv[t] - lse;  // -inf stays -inf
        if (lv[t] > bestv) { bestv = lv[t]; besti = n; }
      }
      for (int off = 16; off > 0; off >>= 1) {
        const float ov = __shfl_xor(bestv, off, 32);
        const int   oi = __shfl_xor(besti, off, 32);
        if (ov > bestv || (ov == bestv && oi < besti)) { bestv = ov; besti = oi; }
      }
      if (lane == 0) {
        const int sel = besti;
        vis[r][sel >> 5] |= (1u << (sel & 31));
        prevL[r] = sel;
        if (step == 0) firstL[r] = sel;
        out_seq[(size_t)b * kN + step] = (float)sel;
      }
    }
    __syncthreads();   // state updates visible before next step's stage 1
  }
}

// ---------------------------------------------------------------------------
extern "C" void kernel_launch(void* const* d_in, const int* in_sizes, int n_in,
                              void* d_out, int out_size, void* d_ws, size_t ws_size,
                              hipStream_t stream) {
  (void)in_sizes; (void)n_in; (void)out_size; (void)ws_size;
  const float* h  = (const float*)d_in[0];
  const float* Wp = (const float*)d_in[1];
  const float* Wn = (const float*)d_in[2];
  const float* Wf = (const float*)d_in[3];
  const float* Ws = (const float*)d_in[4];
  const float* Wo = (const float*)d_in[5];

  // workspace layout (only precompute products; decode state lives in LDS)
  float* qkv    = (float*)d_ws;                         // [B*N][384]
  float* fixedc = qkv + (size_t)kB * kN * kTD;          // [B][128]

  float* out_logp = (float*)d_out;                      // [B][step][N]
  float* out_seq  = out_logp + (size_t)kB * kN * kN;    // [B][N]

  qkv_kernel   <<<(kB * kN) / 16, 128, 0, stream>>>(h, Wn, qkv);
  fixed_kernel <<<kB, 128, 0, stream>>>(h, Wf, fixedc);
  decode_kernel<<<kB / kBPG, 512, 0, stream>>>(h, Wp, Ws, Wo, qkv, fixedc,
                                               out_logp, out_seq);
}